// XGate_549755814194
// MI455X (gfx1250) — compile-verified
//
#include <hip/hip_runtime.h>

// y[j] = x[j ^ 0x841041]  (bit-flip permutation on bits 23,18,12,6,0)
// Complex64 output interleaved as float pairs: out[2j]=re, out[2j+1]=im.
//
// Each thread handles 4 consecutive complex elements j in [4t, 4t+4).
// With mask bit0 set and bits1..5 clear, sources for that group are the
// 4-aligned group s = (4t) ^ 0x841040, read as float4 and swapped in pairs.

#define XMASK_NO_B0 0x841040u
#define LOG2N 24

typedef float float4v __attribute__((ext_vector_type(4)));
typedef int   int4v   __attribute__((ext_vector_type(4)));

#if defined(__gfx1250__) && __has_builtin(__builtin_amdgcn_global_load_async_to_lds_b128)
#define USE_ASYNC_LDS 1
#else
#define USE_ASYNC_LDS 0
#endif

__global__ __launch_bounds__(256) void xgate_permute_kernel(
    const float* __restrict__ xr,
    const float* __restrict__ xi,
    float* __restrict__ out) {
  const unsigned tid = threadIdx.x;
  const unsigned t   = blockIdx.x * 256u + tid;   // one thread per 4 complex elems
  const unsigned j0  = t << 2;                    // first output index of group
  const unsigned s   = j0 ^ XMASK_NO_B0;          // 4-aligned source group

  float4v r, im;

#if USE_ASYNC_LDS
  // CDNA5 async data-mover path: global -> LDS (bypasses VGPRs on the load),
  // tracked by ASYNCcnt; read back with ds_load_b128.
  __shared__ __attribute__((aligned(16))) float lr[256 * 4];
  __shared__ __attribute__((aligned(16))) float li[256 * 4];

  __builtin_amdgcn_global_load_async_to_lds_b128(
      (__attribute__((address_space(1))) int4v*)(xr + s),
      (__attribute__((address_space(3))) int4v*)(&lr[tid * 4]),
      /*offset=*/0, /*cpol=*/0);
  __builtin_amdgcn_global_load_async_to_lds_b128(
      (__attribute__((address_space(1))) int4v*)(xi + s),
      (__attribute__((address_space(3))) int4v*)(&li[tid * 4]),
      /*offset=*/0, /*cpol=*/0);

#if __has_builtin(__builtin_amdgcn_s_wait_asynccnt)
  __builtin_amdgcn_s_wait_asynccnt(0);
#else
  asm volatile("s_wait_asynccnt 0" ::: "memory");
#endif

  r  = *(const float4v*)(&lr[tid * 4]);   // ds_load_b128
  im = *(const float4v*)(&li[tid * 4]);   // ds_load_b128
#else
  // Fallback: direct non-temporal B128 loads (still fully coalesced).
  r  = __builtin_nontemporal_load((const float4v*)(xr + s));
  im = __builtin_nontemporal_load((const float4v*)(xi + s));
#endif

  // Sources for j0..j0+3 are s+1, s, s+3, s+2  -> swap pairs while interleaving.
  float4v o0 = { r.y, im.y, r.x, im.x };  // out[2*j0 .. 2*j0+3]
  float4v o1 = { r.w, im.w, r.z, im.z };  // out[2*j0+4 .. 2*j0+7]

  // Streaming 128 MB written exactly once: non-temporal so we don't churn L2.
  float4v* o = (float4v*)(out + (size_t)t * 8u);
  __builtin_nontemporal_store(o0, o);
  __builtin_nontemporal_store(o1, o + 1);
}

extern "C" void kernel_launch(void* const* d_in, const int* in_sizes, int n_in,
                              void* d_out, int out_size, void* d_ws, size_t ws_size,
                              hipStream_t stream) {
  const float* xr = (const float*)d_in[0];   // x_real, N floats
  const float* xi = (const float*)d_in[1];   // x_imag, N floats
  float* out = (float*)d_out;                // complex64 -> 2N interleaved floats

  const unsigned N = (n_in > 0 && in_sizes[0] > 0) ? (unsigned)in_sizes[0]
                                                   : (1u << LOG2N);
  const unsigned threads = N >> 2;           // 4 complex elems per thread
  const unsigned block = 256;
  const unsigned grid = threads / block;     // N=2^24 -> 16384 blocks

  hipLaunchKernelGGL(xgate_permute_kernel, dim3(grid), dim3(block), 0, stream,
                     xr, xi, out);
}